// CDLoss_51247549775994
// MI455X (gfx1250) — compile-verified
//
#include <hip/hip_runtime.h>

typedef __attribute__((ext_vector_type(2))) float v2f;
typedef __attribute__((ext_vector_type(8))) float v8f;

#define B_BATCH     4
#define NPTS        8192
#define NTILES      (NPTS / 16)              // 512 tiles per cloud per batch
#define PREP_FLOATS (B_BATCH * NPTS * 4)     // 131072 floats per prep array
#define PART_OFF    (2 * PREP_FLOATS)        // partials start here (floats)
#define N_PP_PART   2048                     // prep waves (65536/32)
#define N_MIN_PART  4096                     // 2 dirs * 2048 waves
#define N_PART      (N_PP_PART + N_MIN_PART)
#define BIGF        3.402823466e38f

// Pass 1: prep[c][i] = (p0, p1, p2, p.p) as aligned float4; per-wave sum of p.p.
__global__ void cd_prep_kernel(const float* __restrict__ x,
                               const float* __restrict__ y,
                               float* __restrict__ ws) {
    int idx = blockIdx.x * blockDim.x + threadIdx.x;   // 0 .. 65535
    int c   = idx >> 15;                               // 0 = gen(x), 1 = train(y)
    int r   = idx & 32767;                             // b*8192 + point
    const float* src = c ? y : x;
    float p0 = src[3 * r + 0];
    float p1 = src[3 * r + 1];
    float p2 = src[3 * r + 2];
    float pp = p0 * p0 + p1 * p1 + p2 * p2;
    float4* prep = (float4*)(ws + (size_t)c * PREP_FLOATS);
    prep[r] = make_float4(p0, p1, p2, pp);
    // deterministic per-wave reduction of pp
    float s = pp;
    s += __shfl_xor(s, 16, 32);
    s += __shfl_xor(s, 8, 32);
    s += __shfl_xor(s, 4, 32);
    s += __shfl_xor(s, 2, 32);
    s += __shfl_xor(s, 1, 32);
    if ((threadIdx.x & 31) == 0)
        ws[PART_OFF + (idx >> 5)] = s;
}

// Pass 2: one wave owns one 16-point query tile; loop over all 512 ref tiles.
// WMMA computes C[i,j] = yy[j] - 2*x_i.y_j (yy folded into K=4 via prep layout).
// acc = elementwise min over ref tiles; then per-row lane-min; partial sum out.
__global__ void cd_min_kernel(const float* __restrict__ qprep,
                              const float* __restrict__ rprep,
                              float* __restrict__ part) {
    const int lane = threadIdx.x & 31;
    const int wave = threadIdx.x >> 5;
    const int b    = blockIdx.x >> 6;                     // 64 blocks per batch
    const int qt   = ((blockIdx.x & 63) << 3) + wave;     // query tile 0..511
    const int half = lane >> 4;                           // 0: K0/K1, 1: K2/K3
    const int n16  = lane & 15;

    // A operand: lane<16 -> (-2p0,-2p1); lane>=16 -> (-2p2, 1)
    const float2* qp = (const float2*)(qprep + (size_t)(b * NPTS + qt * 16 + n16) * 4) + half;
    float2 aq = *qp;
    v2f A;
    A.x = -2.0f * aq.x;
    A.y = half ? 1.0f : (-2.0f * aq.y);

    const v8f czero = {0.f, 0.f, 0.f, 0.f, 0.f, 0.f, 0.f, 0.f};
    v8f acc = {BIGF, BIGF, BIGF, BIGF, BIGF, BIGF, BIGF, BIGF};

    // B operand base for this batch: lane<16 -> (y0,y1); lane>=16 -> (y2,yy)
    const float2* rbase = (const float2*)(rprep + (size_t)(b * NPTS + n16) * 4) + half;
    // tile stride: 16 points * 4 floats = 32 float2 (256 bytes)
#pragma unroll 4
    for (int t = 0; t < NTILES; ++t) {
        float2 rb = rbase[(size_t)t * 32];
        v2f Bv;
        Bv.x = rb.x;
        Bv.y = rb.y;
        // (neg_a, A, neg_b, B, c_mod, C, reuse_a, reuse_b)
        v8f c = __builtin_amdgcn_wmma_f32_16x16x4_f32(
            false, A, false, Bv, (short)0, czero, false, false);
#pragma unroll
        for (int i = 0; i < 8; ++i)
            acc[i] = fminf(acc[i], c[i]);
    }

    // Row-min: row i = r + 8*half lives in 16 lanes of VGPR r; min across them.
    float wsum = 0.0f;
#pragma unroll
    for (int r = 0; r < 8; ++r) {
        float m = acc[r];
        m = fminf(m, __shfl_xor(m, 1, 32));
        m = fminf(m, __shfl_xor(m, 2, 32));
        m = fminf(m, __shfl_xor(m, 4, 32));
        m = fminf(m, __shfl_xor(m, 8, 32));
        if (n16 == 0) wsum += m;   // lanes 0 and 16 hold the two row-halves
    }
    wsum += __shfl_xor(wsum, 16, 32);
    if (lane == 0)
        part[blockIdx.x * 8 + wave] = wsum;
}

// Pass 3: fixed-order tree reduction of all partials -> scalar loss.
__global__ void cd_final_kernel(const float* __restrict__ part,
                                float* __restrict__ out) {
    __shared__ float sm[256];
    float s = 0.0f;
    for (int i = threadIdx.x; i < N_PART; i += 256)
        s += part[i];
    sm[threadIdx.x] = s;
    __syncthreads();
    for (int stride = 128; stride > 0; stride >>= 1) {
        if ((int)threadIdx.x < stride)
            sm[threadIdx.x] += sm[threadIdx.x + stride];
        __syncthreads();
    }
    if (threadIdx.x == 0)
        out[0] = sm[0] * (0.5f / (float)B_BATCH);
}

extern "C" void kernel_launch(void* const* d_in, const int* in_sizes, int n_in,
                              void* d_out, int out_size, void* d_ws, size_t ws_size,
                              hipStream_t stream) {
    const float* x = (const float*)d_in[0];   // gen_points_batch   [4,8192,3]
    const float* y = (const float*)d_in[1];   // train_points_dense [4,8192,3]
    float* ws  = (float*)d_ws;
    float* out = (float*)d_out;

    float* prepX = ws;
    float* prepY = ws + PREP_FLOATS;
    float* part  = ws + PART_OFF;             // [pp partials | min partials]

    // 65536 points total (2 clouds * 4 batches * 8192)
    cd_prep_kernel<<<256, 256, 0, stream>>>(x, y, ws);

    // dir 0: queries = x, refs = y ; dir 1: swapped
    cd_min_kernel<<<B_BATCH * 64, 256, 0, stream>>>(prepX, prepY, part + N_PP_PART);
    cd_min_kernel<<<B_BATCH * 64, 256, 0, stream>>>(prepY, prepX, part + N_PP_PART + 2048);

    cd_final_kernel<<<1, 256, 0, stream>>>(part, out);
}